// STELLAR_32057635897573
// MI455X (gfx1250) — compile-verified
//
#include <hip/hip_runtime.h>
#include <hip/hip_bf16.h>

typedef __attribute__((ext_vector_type(2))) float v2f;
typedef __attribute__((ext_vector_type(8))) float v8f;

#define HID 128

// ---------------------------------------------------------------------------
// Dense GEMM tile kernel: Y[16-row tile] = act(X @ W + bias)
//   X: [nrows, K] row-major, W: [K, 128] row-major, bias: [128]
//   One 128-thread block (4 waves) per 16-row tile.
//   Each wave computes two 16x16 output tiles via V_WMMA_F32_16X16X4_F32.
// ---------------------------------------------------------------------------
template <int K, bool RELU>
__global__ void __launch_bounds__(128)
STELLAR_fc_wmma(const float* __restrict__ X, const float* __restrict__ W,
                const float* __restrict__ bias, float* __restrict__ Y,
                int nrows) {
  constexpr int KP = K + 4;                 // LDS row pad: bank-conflict-free
  __shared__ float xs[16 * KP];

  const int row0 = blockIdx.x * 16;
  const int tid  = threadIdx.x;
  const int lane = tid & 31;
  const int wave = tid >> 5;

  // ---- cooperative staging of the 16 x K input tile into LDS -------------
  constexpr int NV4 = (16 * K) / 4;         // float4 elements in tile
  if (row0 + 16 <= nrows) {
    // 16 consecutive rows => one contiguous region of X
    const float4* src = (const float4*)(X + (size_t)row0 * K);
#pragma unroll
    for (int i = tid; i < NV4; i += 128) {
      float4 v = src[i];
      const int li = i * 4;
      const int r = li / K, c = li % K;
      *(float4*)&xs[r * KP + c] = v;
    }
  } else {
    for (int i = tid; i < NV4; i += 128) {
      const int li = i * 4;
      const int r = li / K, c = li % K;
      float4 v = {0.f, 0.f, 0.f, 0.f};
      if (row0 + r < nrows) v = *(const float4*)(X + (size_t)(row0 + r) * K + c);
      *(float4*)&xs[r * KP + c] = v;
    }
  }
  __syncthreads();

  // ---- WMMA fragment coordinates (wave32, 16x16x4 f32) -------------------
  // A (16x4): lanes 0-15 -> K={0,1}, lanes 16-31 -> K={2,3}; M = lane&15
  // B (4x16): mirrored;   N = lane&15
  const int m  = lane & 15;
  const int kg = (lane >> 4) * 2;

#pragma unroll
  for (int t = 0; t < 2; ++t) {
    const int n0 = (wave + 4 * t) * 16;     // this wave's N tile
    const int nn = n0 + m;

    v8f acc;
    const float bv = bias[nn];              // bias is per-column -> all VGPRs
#pragma unroll
    for (int v = 0; v < 8; ++v) acc[v] = bv;

#pragma unroll 8
    for (int k = 0; k < K; k += 4) {
      v2f a;
      a.x = xs[m * KP + k + kg];
      a.y = xs[m * KP + k + kg + 1];
      const float* wp = W + (size_t)(k + kg) * HID + nn;
      if ((k & 31) == 0 && k + 32 < K)
        __builtin_prefetch(wp + 32 * HID, 0, 0);   // global_prefetch_b8
      v2f b;
      b.x = wp[0];
      b.y = wp[HID];
      // D = A(16x4,f32) * B(4x16,f32) + C
      acc = __builtin_amdgcn_wmma_f32_16x16x4_f32(
          false, a, false, b, (short)0, acc, false, false);
    }

    // C/D layout: VGPR v, lanes 0-15 -> M=v ; lanes 16-31 -> M=v+8
    const int rb = row0 + (lane >> 4) * 8;
#pragma unroll
    for (int v = 0; v < 8; ++v) {
      const int r = rb + v;
      float val = acc[v];
      if (RELU) val = fmaxf(val, 0.f);
      if (r < nrows) Y[(size_t)r * HID + nn] = val;
    }
  }
}

// ---------------------------------------------------------------------------
// Fused final GEMM + row softmax:  out = softmax(X @ W + bias, axis=1)
// Same WMMA tiling; logits land in LDS; per-row reduce with wave32 shuffles.
// ---------------------------------------------------------------------------
__global__ void __launch_bounds__(128)
STELLAR_fc_softmax(const float* __restrict__ X, const float* __restrict__ W,
                   const float* __restrict__ bias, float* __restrict__ out,
                   int nrows) {
  constexpr int K = 128, KP = K + 4;
  __shared__ float xs[16 * KP];
  __shared__ float ls[16 * KP];             // logits tile

  const int row0 = blockIdx.x * 16;
  const int tid  = threadIdx.x;
  const int lane = tid & 31;
  const int wave = tid >> 5;

  constexpr int NV4 = (16 * K) / 4;
  if (row0 + 16 <= nrows) {
    const float4* src = (const float4*)(X + (size_t)row0 * K);
#pragma unroll
    for (int i = tid; i < NV4; i += 128) {
      float4 v = src[i];
      const int li = i * 4;
      *(float4*)&xs[(li / K) * KP + (li % K)] = v;
    }
  } else {
    for (int i = tid; i < NV4; i += 128) {
      const int li = i * 4;
      const int r = li / K, c = li % K;
      float4 v = {0.f, 0.f, 0.f, 0.f};
      if (row0 + r < nrows) v = *(const float4*)(X + (size_t)(row0 + r) * K + c);
      *(float4*)&xs[r * KP + c] = v;
    }
  }
  __syncthreads();

  const int m  = lane & 15;
  const int kg = (lane >> 4) * 2;

#pragma unroll
  for (int t = 0; t < 2; ++t) {
    const int n0 = (wave + 4 * t) * 16;
    const int nn = n0 + m;
    v8f acc;
    const float bv = bias[nn];
#pragma unroll
    for (int v = 0; v < 8; ++v) acc[v] = bv;
#pragma unroll 8
    for (int k = 0; k < K; k += 4) {
      v2f a;
      a.x = xs[m * KP + k + kg];
      a.y = xs[m * KP + k + kg + 1];
      const float* wp = W + (size_t)(k + kg) * HID + nn;
      v2f b;
      b.x = wp[0];
      b.y = wp[HID];
      acc = __builtin_amdgcn_wmma_f32_16x16x4_f32(
          false, a, false, b, (short)0, acc, false, false);
    }
    const int rb = (lane >> 4) * 8;
#pragma unroll
    for (int v = 0; v < 8; ++v) ls[(rb + v) * KP + nn] = acc[v];
  }
  __syncthreads();

  // ---- softmax: wave w handles rows 4w..4w+3; 32 lanes x 4 cols each -----
#pragma unroll
  for (int rr = 0; rr < 4; ++rr) {
    const int r = wave * 4 + rr;
    const int grow = row0 + r;
    if (grow >= nrows) continue;            // uniform per wave
    const float4 xv = *(const float4*)&ls[r * KP + lane * 4];
    float mx = fmaxf(fmaxf(xv.x, xv.y), fmaxf(xv.z, xv.w));
#pragma unroll
    for (int o = 16; o > 0; o >>= 1) mx = fmaxf(mx, __shfl_xor(mx, o, 32));
    const float e0 = expf(xv.x - mx), e1 = expf(xv.y - mx);
    const float e2 = expf(xv.z - mx), e3 = expf(xv.w - mx);
    float s = e0 + e1 + e2 + e3;
#pragma unroll
    for (int o = 16; o > 0; o >>= 1) s += __shfl_xor(s, o, 32);
    const float inv = 1.0f / s;
    float4 ov = {e0 * inv, e1 * inv, e2 * inv, e3 * inv};
    *(float4*)&out[(size_t)grow * HID + lane * 4] = ov;
  }
}

// ---------------------------------------------------------------------------
// SpMM scatter: one wave per edge; float4 gather of h[col], 4 HW f32 atomics.
// h and agg (25.6 MB each) are L2-resident on MI455X (192 MB L2).
// ---------------------------------------------------------------------------
__global__ void __launch_bounds__(256)
STELLAR_spmm(const float* __restrict__ vals, const int* __restrict__ rows,
             const int* __restrict__ cols, const float* __restrict__ h,
             float* __restrict__ agg, int nedges) {
  const int gt = blockIdx.x * 256 + threadIdx.x;
  const int e = gt >> 5;
  const int lane = gt & 31;
  if (e >= nedges) return;
  const float v = vals[e];
  const int r = rows[e];
  const int c = cols[e];
  const float4 hv = *(const float4*)(h + (size_t)c * HID + lane * 4);
  float* ap = agg + (size_t)r * HID + lane * 4;
  unsafeAtomicAdd(ap + 0, v * hv.x);
  unsafeAtomicAdd(ap + 1, v * hv.y);
  unsafeAtomicAdd(ap + 2, v * hv.z);
  unsafeAtomicAdd(ap + 3, v * hv.w);
}

__global__ void __launch_bounds__(256)
STELLAR_zero(float4* __restrict__ p, int n4) {
  const int i = blockIdx.x * 256 + threadIdx.x;
  if (i < n4) {
    float4 z = {0.f, 0.f, 0.f, 0.f};
    p[i] = z;
  }
}

// ---------------------------------------------------------------------------
extern "C" void kernel_launch(void* const* d_in, const int* in_sizes, int n_in,
                              void* d_out, int out_size, void* d_ws,
                              size_t ws_size, hipStream_t stream) {
  const float* x   = (const float*)d_in[0];   // [N, 256]
  const float* W1  = (const float*)d_in[1];   // [256, 128]
  const float* b1  = (const float*)d_in[2];   // [128]
  const float* W2  = (const float*)d_in[3];   // [128, 128]
  const float* b2  = (const float*)d_in[4];   // [128]
  const float* av  = (const float*)d_in[5];   // [E]
  const int*   ar  = (const int*)d_in[6];     // [E]
  const int*   ac  = (const int*)d_in[7];     // [E]
  float* out = (float*)d_out;

  const int N = in_sizes[0] / 256;
  const int E = in_sizes[5];

  float* h   = (float*)d_ws;                  // [N,128]; later reused as h2
  float* agg = h + (size_t)N * HID;           // [N,128]

  const int nb = (N + 15) / 16;

  // 1) h = relu(x @ W1 + b1)
  STELLAR_fc_wmma<256, true><<<nb, 128, 0, stream>>>(x, W1, b1, h, N);

  // 2) agg = 0 ; agg[row] += val * h[col]
  const int n4 = (N * HID) / 4;
  STELLAR_zero<<<(n4 + 255) / 256, 256, 0, stream>>>((float4*)agg, n4);
  const long long ethreads = (long long)E * 32;
  STELLAR_spmm<<<(int)((ethreads + 255) / 256), 256, 0, stream>>>(
      av, ar, ac, h, agg, E);

  // 3) h2 = agg @ W2 + b2   (reuse h buffer)
  STELLAR_fc_wmma<128, false><<<nb, 128, 0, stream>>>(agg, W2, b2, h, N);

  // 4) out = softmax(h2 @ W2 + b2)
  STELLAR_fc_softmax<<<nb, 128, 0, stream>>>(h, W2, b2, out, N);
}